// SS2D_13219909337454
// MI455X (gfx1250) — compile-verified
//
#include <hip/hip_runtime.h>
#include <math.h>

// ---------------- problem constants ----------------
#define BATCH   2
#define HH      48
#define WW      48
#define DMODEL  96
#define DIN     192
#define NST     16
#define DTR     6
#define LQ      (HH*WW)          // 2304
#define BL      (BATCH*LQ)       // 4608
#define KDIRS   4
#define BKL     (BATCH*KDIRS*LQ) // 18432
#define XCOLS   (DTR + 2*NST)    // 38

typedef __attribute__((ext_vector_type(2))) float v2f;
typedef __attribute__((ext_vector_type(8))) float v8f;

// scan-direction position map: scan index l -> row-major spatial index p
__device__ __forceinline__ int dir_pos(int kdir, int l) {
    if (kdir == 0) return l;
    if (kdir == 1) return (l % HH) * WW + (l / HH);
    if (kdir == 2) return LQ - 1 - l;
    int l2 = LQ - 1 - l;
    return (l2 % HH) * WW + (l2 / HH);
}

// ---------------- 1) in_proj: (BL x 96) @ (96 x 384) -> xz ----------------
__global__ void k_inproj(const float* __restrict__ X, const float* __restrict__ Win,
                         float* __restrict__ XZ) {
    const int wave = blockIdx.x * (blockDim.x >> 5) + (threadIdx.x >> 5);
    const int lane = threadIdx.x & 31;
    const int half = lane >> 4, l16 = lane & 15;
    const int NT = (2*DIN) / 16;                 // 24
    const int n0 = (wave % NT) * 16;
    const int m0 = (wave / NT) * 16;
    const float* arow = X + (m0 + l16) * DMODEL;
    v8f acc = {};
    for (int k = 0; k < DMODEL; k += 4) {
        v2f a, b;
        a.x = arow[k + 2*half];
        a.y = arow[k + 1 + 2*half];
        b.x = Win[(k + 2*half)     * (2*DIN) + n0 + l16];
        b.y = Win[(k + 1 + 2*half) * (2*DIN) + n0 + l16];
        acc = __builtin_amdgcn_wmma_f32_16x16x4_f32(false, a, false, b,
                                                    (short)0, acc, false, false);
    }
    #pragma unroll
    for (int r = 0; r < 8; ++r)
        XZ[(m0 + r + 8*half) * (2*DIN) + n0 + l16] = acc[r];
}

// ---------------- 2) depthwise 3x3 conv + SiLU ----------------
__global__ void k_conv(const float* __restrict__ XZ, const float* __restrict__ CW,
                       const float* __restrict__ CB, float* __restrict__ XCS) {
    int idx = blockIdx.x * blockDim.x + threadIdx.x;
    if (idx >= BL * DIN) return;
    const int d  = idx % DIN;
    const int pl = idx / DIN;
    const int b  = pl / LQ;
    const int l  = pl % LQ;
    const int h  = l / WW, w = l % WW;
    float acc = CB[d];
    #pragma unroll
    for (int ky = 0; ky < 3; ++ky) {
        int hh = h + ky - 1;
        if (hh < 0 || hh >= HH) continue;
        #pragma unroll
        for (int kx = 0; kx < 3; ++kx) {
            int ww = w + kx - 1;
            if (ww < 0 || ww >= WW) continue;
            float xv = XZ[((b * LQ + hh * WW + ww) * (2*DIN)) + d]; // xp part
            acc = fmaf(xv, CW[d * 9 + ky * 3 + kx], acc);
        }
    }
    XCS[idx] = acc / (1.f + expf(-acc));          // SiLU
}

// ---------------- 3) x_proj with gathered rows: (BKL x 192) @ (192 x 38) ----------------
__global__ void k_xproj(const float* __restrict__ XCS, const float* __restrict__ Wx,
                        float* __restrict__ XDBL) {
    const int wave = blockIdx.x * (blockDim.x >> 5) + (threadIdx.x >> 5);
    const int lane = threadIdx.x & 31;
    const int half = lane >> 4, l16 = lane & 15;
    const int NT = 3;                              // covers 48 cols, guard < 38
    const int n0 = (wave % NT) * 16;
    const int m0 = (wave / NT) * 16;
    // gather: global row -> (b, kdir, l) -> spatial p
    const int r    = m0 + l16;
    const int b    = r / (KDIRS * LQ);
    const int rem  = r % (KDIRS * LQ);
    const int kdir = rem / LQ;
    const int l    = rem % LQ;
    const float* arow = XCS + (b * LQ + dir_pos(kdir, l)) * DIN;
    const int col = n0 + l16;
    const bool cok = (col < XCOLS);
    v8f acc = {};
    for (int k = 0; k < DIN; k += 4) {
        v2f a, bf;
        a.x = arow[k + 2*half];
        a.y = arow[k + 1 + 2*half];
        bf.x = cok ? Wx[(k + 2*half)     * XCOLS + col] : 0.f;
        bf.y = cok ? Wx[(k + 1 + 2*half) * XCOLS + col] : 0.f;
        acc = __builtin_amdgcn_wmma_f32_16x16x4_f32(false, a, false, bf,
                                                    (short)0, acc, false, false);
    }
    if (cok) {
        #pragma unroll
        for (int rr = 0; rr < 8; ++rr)
            XDBL[(m0 + rr + 8*half) * XCOLS + col] = acc[rr];
    }
}

// ---------------- 4) dt_proj (K=6 padded to 8) + bias + softplus ----------------
__global__ void k_dtproj(const float* __restrict__ XDBL, const float* __restrict__ Wdt,
                         const float* __restrict__ Bdt, float* __restrict__ DSP) {
    const int wave = blockIdx.x * (blockDim.x >> 5) + (threadIdx.x >> 5);
    const int lane = threadIdx.x & 31;
    const int half = lane >> 4, l16 = lane & 15;
    const int NT = DIN / 16;                       // 12
    const int n0 = (wave % NT) * 16;
    const int m0 = (wave / NT) * 16;
    const float* arow = XDBL + (m0 + l16) * XCOLS; // dt = cols 0..5
    const int col = n0 + l16;
    v8f acc = {};
    #pragma unroll
    for (int k = 0; k < 8; k += 4) {
        int k0 = k + 2*half, k1 = k + 1 + 2*half;
        v2f a, bf;
        a.x  = (k0 < DTR) ? arow[k0] : 0.f;
        a.y  = (k1 < DTR) ? arow[k1] : 0.f;
        bf.x = (k0 < DTR) ? Wdt[k0 * DIN + col] : 0.f;
        bf.y = (k1 < DTR) ? Wdt[k1 * DIN + col] : 0.f;
        acc = __builtin_amdgcn_wmma_f32_16x16x4_f32(false, a, false, bf,
                                                    (short)0, acc, false, false);
    }
    const float bias = Bdt[col];
    #pragma unroll
    for (int rr = 0; rr < 8; ++rr) {
        float v = acc[rr] + bias;
        float sp = (v > 20.f) ? v : log1pf(expf(v));   // softplus
        DSP[(m0 + rr + 8*half) * DIN + col] = sp;
    }
}

// ---------------- 5) selective scan: one (b,kdir,d) chain per 16-lane group ----------------
// Deterministic: each chain owns its (b,kdir,:,d) slice of YS — no atomics.
__global__ void k_scan(const float* __restrict__ XCS, const float* __restrict__ XDBL,
                       const float* __restrict__ DSP, const float* __restrict__ Alog,
                       const float* __restrict__ Dp, float* __restrict__ YS) {
    const int gid = blockIdx.x * (blockDim.x >> 4) + (threadIdx.x >> 4);
    const int n   = threadIdx.x & 15;              // state index
    const int d    = gid % DIN;
    const int kdir = (gid / DIN) % KDIRS;
    const int b    = gid / (DIN * KDIRS);
    const float Ad = -expf(Alog[d * NST + n]);
    const float Dv = Dp[d];
    const int rbase = (b * KDIRS + kdir) * LQ;
    float h = 0.f;
    for (int l = 0; l < LQ; ++l) {
        const int row = rbase + l;
        if (l + 8 < LQ)
            __builtin_prefetch(&XDBL[(row + 8) * XCOLS], 0, 1);
        const float delta = DSP[row * DIN + d];
        const int p = dir_pos(kdir, l);
        const float u  = XCS[(b * LQ + p) * DIN + d];
        const float Bv = XDBL[row * XCOLS + DTR + n];
        const float Cv = XDBL[row * XCOLS + DTR + NST + n];
        h = expf(delta * Ad) * h + (delta * u) * Bv;
        float t = h * Cv;
        t += __shfl_xor(t, 8, 32);
        t += __shfl_xor(t, 4, 32);
        t += __shfl_xor(t, 2, 32);
        t += __shfl_xor(t, 1, 32);
        if (n == 0)                                 // faithful: store at scan slot l
            YS[row * DIN + d] = t + Dv * u;
    }
}

// ---------------- 6) 4-direction sum + LayerNorm + SiLU(z) gate: one wave per row ----------------
__global__ void k_lngate(const float* __restrict__ YS, const float* __restrict__ XZ,
                         const float* __restrict__ G, const float* __restrict__ Bt,
                         float* __restrict__ YG) {
    const int row  = blockIdx.x * (blockDim.x >> 5) + (threadIdx.x >> 5); // b*LQ + l
    const int lane = threadIdx.x & 31;
    const int b = row / LQ;
    const int l = row % LQ;
    const int base = (b * KDIRS) * LQ + l;          // YS row index for kdir=0
    float v[6], s = 0.f, ss = 0.f;
    #pragma unroll
    for (int j = 0; j < 6; ++j) {
        const int c = lane + 32 * j;
        float acc = 0.f;                             // deterministic 4-way sum
        #pragma unroll
        for (int k = 0; k < KDIRS; ++k)
            acc += YS[(base + k * LQ) * DIN + c];
        v[j] = acc;
        s += acc; ss += acc * acc;
    }
    #pragma unroll
    for (int off = 16; off; off >>= 1) {
        s  += __shfl_xor(s,  off, 32);
        ss += __shfl_xor(ss, off, 32);
    }
    const float mu  = s  * (1.f / DIN);
    const float var = ss * (1.f / DIN) - mu * mu;
    const float rs  = rsqrtf(var + 1e-5f);
    #pragma unroll
    for (int j = 0; j < 6; ++j) {
        const int c = lane + 32 * j;
        const float zv = XZ[row * (2*DIN) + DIN + c];
        const float gate = zv / (1.f + expf(-zv));
        YG[row * DIN + c] = ((v[j] - mu) * rs * G[c] + Bt[c]) * gate;
    }
}

// ---------------- 7) out_proj: (BL x 192) @ (192 x 96) -> out ----------------
__global__ void k_outproj(const float* __restrict__ YG, const float* __restrict__ Wout,
                          float* __restrict__ OUT) {
    const int wave = blockIdx.x * (blockDim.x >> 5) + (threadIdx.x >> 5);
    const int lane = threadIdx.x & 31;
    const int half = lane >> 4, l16 = lane & 15;
    const int NT = DMODEL / 16;                    // 6
    const int n0 = (wave % NT) * 16;
    const int m0 = (wave / NT) * 16;
    const float* arow = YG + (m0 + l16) * DIN;
    v8f acc = {};
    for (int k = 0; k < DIN; k += 4) {
        v2f a, b;
        a.x = arow[k + 2*half];
        a.y = arow[k + 1 + 2*half];
        b.x = Wout[(k + 2*half)     * DMODEL + n0 + l16];
        b.y = Wout[(k + 1 + 2*half) * DMODEL + n0 + l16];
        acc = __builtin_amdgcn_wmma_f32_16x16x4_f32(false, a, false, b,
                                                    (short)0, acc, false, false);
    }
    #pragma unroll
    for (int r = 0; r < 8; ++r)
        OUT[(m0 + r + 8*half) * DMODEL + n0 + l16] = acc[r];
}

extern "C" void kernel_launch(void* const* d_in, const int* in_sizes, int n_in,
                              void* d_out, int out_size, void* d_ws, size_t ws_size,
                              hipStream_t stream) {
    (void)in_sizes; (void)n_in; (void)out_size; (void)ws_size;
    const float* x      = (const float*)d_in[0];
    const float* w_in   = (const float*)d_in[1];
    const float* conv_w = (const float*)d_in[2];
    const float* conv_b = (const float*)d_in[3];
    const float* w_x    = (const float*)d_in[4];
    const float* w_dt   = (const float*)d_in[5];
    const float* b_dt   = (const float*)d_in[6];
    const float* A_log  = (const float*)d_in[7];
    const float* Dp     = (const float*)d_in[8];
    const float* ln_g   = (const float*)d_in[9];
    const float* ln_b   = (const float*)d_in[10];
    const float* w_out  = (const float*)d_in[11];
    float* out = (float*)d_out;

    float* ws   = (float*)d_ws;
    float* XZ   = ws;                        // BL * 384
    float* XCS  = XZ   + (size_t)BL  * (2*DIN);
    float* XDBL = XCS  + (size_t)BL  * DIN;  // BKL * 38
    float* DSP  = XDBL + (size_t)BKL * XCOLS;
    float* YS   = DSP  + (size_t)BKL * DIN;  // BKL * 192 (per-direction y)
    float* YG   = YS   + (size_t)BKL * DIN;

    // 1) in_proj: (BL/16)*(384/16) = 6912 waves, 8 waves/block
    k_inproj<<<6912 / 8, 256, 0, stream>>>(x, w_in, XZ);
    // 2) conv + SiLU
    k_conv<<<(BL * DIN + 255) / 256, 256, 0, stream>>>(XZ, conv_w, conv_b, XCS);
    // 3) x_proj: (BKL/16)*3 = 3456 waves
    k_xproj<<<3456 / 8, 256, 0, stream>>>(XCS, w_x, XDBL);
    // 4) dt_proj: (BKL/16)*12 = 13824 waves
    k_dtproj<<<13824 / 8, 256, 0, stream>>>(XDBL, w_dt, b_dt, DSP);
    // 5) scan: 1536 chains, 8 chains per 128-thread block
    k_scan<<<1536 / 8, 128, 0, stream>>>(XCS, XDBL, DSP, A_log, Dp, YS);
    // 6) 4-dir sum + LN + gate: 4 rows per 128-thread block
    k_lngate<<<BL / 4, 128, 0, stream>>>(YS, XZ, ln_g, ln_b, YG);
    // 7) out_proj: (BL/16)*(96/16) = 1728 waves
    k_outproj<<<1728 / 8, 256, 0, stream>>>(YG, w_out, out);
}